// KalmanFilter_15668040695945
// MI455X (gfx1250) — compile-verified
//
#include <hip/hip_runtime.h>

// ---------------------------------------------------------------------------
// Kalman filter one-step-ahead predictions on MI455X (gfx1250).
// One wave32 per group; state P (16x16) and m (16) live in VGPRs in WMMA
// C-layout across the whole T=512 scan. All 16x16 matmuls use
// v_wmma_f32_16x16x4_f32 (the only f32 WMMA shape). Layout conversions and
// transposes are staged through per-wave private LDS tiles (in-order DS,
// no barriers needed). The 4x4 innovation solve is done uniformly per wave
// via shuffles + cofactor inverse.
// ---------------------------------------------------------------------------

typedef __attribute__((ext_vector_type(2))) float v2f;
typedef __attribute__((ext_vector_type(8))) float v8f;

#define G_GROUPS 1024
#define T_STEPS  512
#define WAVES_PER_BLOCK 4
#define THREADS (WAVES_PER_BLOCK * 32)

static __device__ __forceinline__ v8f wmma4(v2f a, v2f b, v8f c) {
  // D(16x16,f32) = A(16x4,f32) * B(4x16,f32) + C
  return __builtin_amdgcn_wmma_f32_16x16x4_f32(
      /*neg_a=*/false, a, /*neg_b=*/false, b,
      /*c_mod=*/(short)0, c, /*reuse_a=*/false, /*reuse_b=*/false);
}

static __device__ __forceinline__ v8f zero8() {
  v8f z = {0.f, 0.f, 0.f, 0.f, 0.f, 0.f, 0.f, 0.f};
  return z;
}

// C/D layout: reg r holds row (r + 8*half), column (lane & 15).
static __device__ __forceinline__ void stage_c(float* buf, v8f c, int col, int half) {
#pragma unroll
  for (int r = 0; r < 8; ++r) buf[(r + 8 * half) * 16 + col] = c[r];
}

// A-operand chunk k of a row-major 16x16 LDS tile:
// lane (m = lane&15) holds elements (m, 4k + 2*half + {0,1}).
static __device__ __forceinline__ v2f load_a(const float* buf, int col, int half, int k) {
  int k0 = 4 * k + 2 * half;
  v2f a;
  a.x = buf[col * 16 + k0];
  a.y = buf[col * 16 + k0 + 1];
  return a;
}

// B-operand chunk k of a row-major 16x16 LDS tile:
// lane (n = lane&15) holds elements (4k + 2*half + {0,1}, n).
static __device__ __forceinline__ v2f load_b(const float* buf, int col, int half, int k) {
  int k0 = 4 * k + 2 * half;
  v2f b;
  b.x = buf[k0 * 16 + col];
  b.y = buf[(k0 + 1) * 16 + col];
  return b;
}

// B-operand chunk k of the TRANSPOSE of a row-major LDS tile (X^T as B).
static __device__ __forceinline__ v2f load_bt(const float* buf, int col, int half, int k) {
  int k0 = 4 * k + 2 * half;
  v2f b;
  b.x = buf[col * 16 + k0];
  b.y = buf[col * 16 + k0 + 1];
  return b;
}

static __device__ __forceinline__ float sel4(int i, float a, float b, float c, float d) {
  return i == 0 ? a : (i == 1 ? b : (i == 2 ? c : d));
}

__global__ __launch_bounds__(THREADS, 1)
void kalman_wmma_kernel(const float* __restrict__ y,   // [G,T,4]
                        const float* __restrict__ F,   // [16,16]
                        const float* __restrict__ H,   // [4,16]
                        const float* __restrict__ Q,   // [16,16]
                        const float* __restrict__ R,   // [4,4]
                        const float* __restrict__ im,  // [G,16]
                        const float* __restrict__ ic,  // [16,16]
                        float* __restrict__ out) {     // [G,T,4]
  __shared__ float lds[WAVES_PER_BLOCK * 4 * 256];

  const int lane = threadIdx.x & 31;
  const int wave = threadIdx.x >> 5;
  const int col  = lane & 15;
  const int half = lane >> 4;
  const int g    = blockIdx.x * WAVES_PER_BLOCK + wave;

  float* bufP = &lds[(wave * 4 + 0) * 256];  // P tile -> T tile -> P_u -> ...
  float* bufZ = &lds[(wave * 4 + 1) * 256];  // [P*H^T | m] tile -> K tile
  float* bufI = &lds[(wave * 4 + 2) * 256];  // IKH tile -> T2 tile
  float* bufS = &lds[(wave * 4 + 3) * 256];  // K*R tile -> m_u column

  // ---- constants in WMMA operand layouts (held in VGPRs) ------------------
  // FA[k]: A-chunk of F, which is numerically identical to the B-chunk of F^T.
  // HA[k]: A-chunk of H padded to 16x16 (rows 4..15 zero) == B-chunk of H^T
  //        padded to 16x16 (cols 4..15 zero).
  v2f FA[4], HA[4];
#pragma unroll
  for (int k = 0; k < 4; ++k) {
    int k0 = 4 * k + 2 * half;
    FA[k].x = F[col * 16 + k0];
    FA[k].y = F[col * 16 + k0 + 1];
    HA[k].x = (col < 4) ? H[col * 16 + k0] : 0.f;
    HA[k].y = (col < 4) ? H[col * 16 + k0 + 1] : 0.f;
  }
  // Hb: B-chunk0 of H itself (4x16), used in IKH = (-K) * H + I.
  v2f Hb;
  Hb.x = H[(2 * half) * 16 + col];
  Hb.y = H[(2 * half + 1) * 16 + col];
  // Rb: B-chunk0 of R padded to 4x16 (cols 4..15 zero), used in KR = K * R.
  v2f Rb;
  Rb.x = (col < 4) ? R[(2 * half) * 4 + col] : 0.f;
  Rb.y = (col < 4) ? R[(2 * half + 1) * 4 + col] : 0.f;

  v8f Qc, Rc, Ic, Pc, Mc;
#pragma unroll
  for (int r = 0; r < 8; ++r) {
    int row = r + 8 * half;
    Qc[r] = Q[row * 16 + col];
    Rc[r] = (row < 4 && col < 4) ? R[row * 4 + col] : 0.f;
    Ic[r] = (row == col) ? 1.f : 0.f;
    Pc[r] = ic[row * 16 + col];                       // init_cov
    Mc[r] = (col == 0) ? im[g * 16 + row] : 0.f;      // init_mean as column-0
  }

  for (int t = 0; t < T_STEPS; ++t) {
    // ---- PHt = P @ H^T (16x16x16: 4 WMMAs; cols 4..15 come out zero) ------
    stage_c(bufP, Pc, col, half);
    v8f PHt = zero8();
#pragma unroll
    for (int k = 0; k < 4; ++k)
      PHt = wmma4(load_a(bufP, col, half, k), HA[k], PHt);

    // ---- Z = [PHt(:,0:3) | m in col 4 | 0]; S&mm = H_f @ Z + R_f ----------
    stage_c(bufZ, PHt, col, half);
    if (col == 0) {
#pragma unroll
      for (int r = 0; r < 8; ++r) bufZ[(r + 8 * half) * 16 + 4] = Mc[r];
    }
    v8f Sc = Rc;
#pragma unroll
    for (int k = 0; k < 4; ++k)
      Sc = wmma4(HA[k], load_b(bufZ, col, half, k), Sc);
    // Sc: rows0..3 x cols0..3 = innovation covariance S; col 4 = H@m (+R col4=0)

    // ---- emit one-step-ahead prediction (lane 4 holds all 4 values) -------
    if (lane == 4) {
      float4 o;
      o.x = Sc[0]; o.y = Sc[1]; o.z = Sc[2]; o.w = Sc[3];
      *(float4*)(out + ((size_t)g * T_STEPS + t) * 4) = o;
    }

    // ---- broadcast S and meas_mean uniformly across the wave --------------
    float s00 = __shfl(Sc[0], 0, 32), s01 = __shfl(Sc[0], 1, 32),
          s02 = __shfl(Sc[0], 2, 32), s03 = __shfl(Sc[0], 3, 32);
    float s10 = __shfl(Sc[1], 0, 32), s11 = __shfl(Sc[1], 1, 32),
          s12 = __shfl(Sc[1], 2, 32), s13 = __shfl(Sc[1], 3, 32);
    float s20 = __shfl(Sc[2], 0, 32), s21 = __shfl(Sc[2], 1, 32),
          s22 = __shfl(Sc[2], 2, 32), s23 = __shfl(Sc[2], 3, 32);
    float s30 = __shfl(Sc[3], 0, 32), s31 = __shfl(Sc[3], 1, 32),
          s32 = __shfl(Sc[3], 2, 32), s33 = __shfl(Sc[3], 3, 32);
    float mm0 = __shfl(Sc[0], 4, 32), mm1 = __shfl(Sc[1], 4, 32),
          mm2 = __shfl(Sc[2], 4, 32), mm3 = __shfl(Sc[3], 4, 32);

    // ---- residual ---------------------------------------------------------
    const float4 yv = *(const float4*)(y + ((size_t)g * T_STEPS + t) * 4);
    float r0 = yv.x - mm0, r1 = yv.y - mm1, r2 = yv.z - mm2, r3 = yv.w - mm3;

    // ---- uniform 4x4 cofactor inverse of S --------------------------------
    float A2323 = s22 * s33 - s23 * s32;
    float A1323 = s21 * s33 - s23 * s31;
    float A1223 = s21 * s32 - s22 * s31;
    float A0323 = s20 * s33 - s23 * s30;
    float A0223 = s20 * s32 - s22 * s30;
    float A0123 = s20 * s31 - s21 * s30;
    float A2313 = s12 * s33 - s13 * s32;
    float A1313 = s11 * s33 - s13 * s31;
    float A1213 = s11 * s32 - s12 * s31;
    float A2312 = s12 * s23 - s13 * s22;
    float A1312 = s11 * s23 - s13 * s21;
    float A1212 = s11 * s22 - s12 * s21;
    float A0313 = s10 * s33 - s13 * s30;
    float A0213 = s10 * s32 - s12 * s30;
    float A0312 = s10 * s23 - s13 * s20;
    float A0212 = s10 * s22 - s12 * s20;
    float A0113 = s10 * s31 - s11 * s30;
    float A0112 = s10 * s21 - s11 * s20;

    float det = s00 * (s11 * A2323 - s12 * A1323 + s13 * A1223)
              - s01 * (s10 * A2323 - s12 * A0323 + s13 * A0223)
              + s02 * (s10 * A1323 - s11 * A0323 + s13 * A0123)
              - s03 * (s10 * A1223 - s11 * A0223 + s12 * A0123);
    float id = 1.0f / det;

    float i00 =  id * (s11 * A2323 - s12 * A1323 + s13 * A1223);
    float i01 = -id * (s01 * A2323 - s02 * A1323 + s03 * A1223);
    float i02 =  id * (s01 * A2313 - s02 * A1313 + s03 * A1213);
    float i03 = -id * (s01 * A2312 - s02 * A1312 + s03 * A1212);
    float i10 = -id * (s10 * A2323 - s12 * A0323 + s13 * A0223);
    float i11 =  id * (s00 * A2323 - s02 * A0323 + s03 * A0223);
    float i12 = -id * (s00 * A2313 - s02 * A0313 + s03 * A0213);
    float i13 =  id * (s00 * A2312 - s02 * A0312 + s03 * A0212);
    float i20 =  id * (s10 * A1323 - s11 * A0323 + s13 * A0123);
    float i21 = -id * (s00 * A1323 - s01 * A0323 + s03 * A0123);
    float i22 =  id * (s00 * A1313 - s01 * A0313 + s03 * A0113);
    float i23 = -id * (s00 * A1312 - s01 * A0312 + s03 * A0112);
    float i30 = -id * (s10 * A1223 - s11 * A0223 + s12 * A0123);
    float i31 =  id * (s00 * A1223 - s01 * A0223 + s02 * A0123);
    float i32 = -id * (s00 * A1213 - s01 * A0213 + s02 * A0113);
    float i33 =  id * (s00 * A1212 - s01 * A0212 + s02 * A0112);

    // ---- K = PHt * Sinv (one WMMA: only K-chunk0 of PHt is nonzero) -------
    float si_r0 = sel4(col, i00, i01, i02, i03);
    float si_r1 = sel4(col, i10, i11, i12, i13);
    float si_r2 = sel4(col, i20, i21, i22, i23);
    float si_r3 = sel4(col, i30, i31, i32, i33);
    v2f Sib;
    Sib.x = (col < 4) ? (half ? si_r2 : si_r0) : 0.f;
    Sib.y = (col < 4) ? (half ? si_r3 : si_r1) : 0.f;

    v2f phta = load_a(bufZ, col, half, 0);         // PHt cols 0..3
    v8f Kc = wmma4(phta, Sib, zero8());

    stage_c(bufZ, Kc, col, half);                  // bufZ now holds K
    v2f ka = load_a(bufZ, col, half, 0);           // K as A-operand (cols 0..3)
    v2f nka; nka.x = -ka.x; nka.y = -ka.y;

    // ---- IKH = I - K*H (one WMMA) -----------------------------------------
    v8f IKH = wmma4(nka, Hb, Ic);

    // ---- m_u = m + K * resid (one WMMA, resid as 4x1 column B) ------------
    v2f rb;
    rb.x = (col == 0) ? (half ? r2 : r0) : 0.f;
    rb.y = (col == 0) ? (half ? r3 : r1) : 0.f;
    v8f Mu = wmma4(ka, rb, Mc);

    // ---- T = IKH @ P (4 WMMAs) --------------------------------------------
    stage_c(bufI, IKH, col, half);
    v8f Tm = zero8();
#pragma unroll
    for (int k = 0; k < 4; ++k)
      Tm = wmma4(load_a(bufI, col, half, k), load_b(bufP, col, half, k), Tm);

    // ---- KRKt = (K*R) @ K^T (two single-chunk WMMAs) ----------------------
    v8f KR = wmma4(ka, Rb, zero8());
    stage_c(bufS, KR, col, half);
    v8f KKt = wmma4(load_a(bufS, col, half, 0), load_bt(bufZ, col, half, 0),
                    zero8());

    // ---- P_u = T @ IKH^T + KRKt (4 WMMAs) ---------------------------------
    stage_c(bufP, Tm, col, half);                  // reuse P tile for T
    v8f Pu = KKt;
#pragma unroll
    for (int k = 0; k < 4; ++k)
      Pu = wmma4(load_a(bufP, col, half, k), load_bt(bufI, col, half, k), Pu);

    // ---- P_n = F @ P_u @ F^T + Q (8 WMMAs) --------------------------------
    stage_c(bufP, Pu, col, half);
    v8f T2 = zero8();
#pragma unroll
    for (int k = 0; k < 4; ++k)
      T2 = wmma4(FA[k], load_b(bufP, col, half, k), T2);
    stage_c(bufI, T2, col, half);
    v8f Pn = Qc;
#pragma unroll
    for (int k = 0; k < 4; ++k)
      Pn = wmma4(load_a(bufI, col, half, k), FA[k] /* == B-chunk of F^T */, Pn);

    // ---- m_n = F @ m_u (4 WMMAs, m_u staged as a 16-float column) ---------
    if (col == 0) {
#pragma unroll
      for (int r = 0; r < 8; ++r) bufS[r + 8 * half] = Mu[r];
    }
    v8f Mn = zero8();
#pragma unroll
    for (int k = 0; k < 4; ++k) {
      float b0 = bufS[4 * k + 2 * half];
      float b1 = bufS[4 * k + 2 * half + 1];
      v2f mb;
      mb.x = (col == 0) ? b0 : 0.f;
      mb.y = (col == 0) ? b1 : 0.f;
      Mn = wmma4(FA[k], mb, Mn);
    }

    Pc = Pn;
    Mc = Mn;
  }
}

extern "C" void kernel_launch(void* const* d_in, const int* in_sizes, int n_in,
                              void* d_out, int out_size, void* d_ws, size_t ws_size,
                              hipStream_t stream) {
  (void)in_sizes; (void)n_in; (void)out_size; (void)d_ws; (void)ws_size;
  const float* y  = (const float*)d_in[0];
  const float* F  = (const float*)d_in[1];
  const float* H  = (const float*)d_in[2];
  const float* Q  = (const float*)d_in[3];
  const float* R  = (const float*)d_in[4];
  const float* im = (const float*)d_in[5];
  const float* ic = (const float*)d_in[6];
  float* out = (float*)d_out;

  dim3 grid(G_GROUPS / WAVES_PER_BLOCK);
  dim3 block(THREADS);
  hipLaunchKernelGGL(kalman_wmma_kernel, grid, block, 0, stream,
                     y, F, H, Q, R, im, ic, out);
}